// AI4DEM_57947698758170
// MI455X (gfx1250) — compile-verified
//
#include <hip/hip_runtime.h>
#include <cmath>
#include <cstdint>

#define DZc 48
#define DYc 256
#define DXc 256
#define NCELL (DZc * DYc * DXc)        // 3,145,728 cells
#define TS 8                            // tile edge (cells)
#define HS (TS + 4)                     // halo edge = 12
#define HC (HS * HS * HS)               // 1728 halo cells
#define NF 9                            // fields per cell
#define HTOT (HC * NF)                  // 15552 floats staged per block
#define KNc 200.0f
#define MUc 0.8f
#define EPSc 1.0e-4f

// ---------------------------------------------------------------------------
// Tiled force/torque kernel (defined FIRST so the disasm snippet shows the
// CDNA5 async global->LDS staging sequence).
//
// Halo staged to LDS with global_load_async_to_lds_b32 (ASYNCcnt path),
// completed with s_wait_asynccnt + workgroup barrier, then the 125-point
// stencil is evaluated entirely out of LDS. Tile = 8^3 cells, halo = 12^3
// cells x 9 fields = 62208 B LDS.
// ---------------------------------------------------------------------------
__global__ void __launch_bounds__(256) dem_forces(
    const float* __restrict__ grid, const int* __restrict__ idxg,
    float* __restrict__ out, int N, float Dc, float two_d, float eta)
{
    __shared__ float tile[HTOT];

    const int t  = threadIdx.x;
    const int tb = blockIdx.x;                       // 6 * 32 * 32 tiles
    const int tz0 = (tb / (32 * 32)) * TS;
    const int ty0 = ((tb / 32) % 32) * TS;
    const int tx0 = (tb % 32) * TS;

    // --- stage halo (with periodic wrap) into LDS via async global->LDS ---
    const uint64_t gbase = (uint64_t)(uintptr_t)grid;
    const uint32_t lds0  = (uint32_t)(uintptr_t)(void*)tile;  // LDS byte offset

    #pragma unroll 1
    for (int it = 0; it < (HTOT + 255) / 256; ++it) {
        int e = it * 256 + t;
        if (e >= HTOT) e = HTOT - 1;                 // tail lanes rewrite last elt
        int f  = e / HC;
        int r  = e - f * HC;
        int hz = r / (HS * HS);
        int hy = (r / HS) % HS;
        int hx = r - hz * (HS * HS) - hy * HS;
        int gz = tz0 - 2 + hz; gz += (gz < 0) ? DZc : 0; gz -= (gz >= DZc) ? DZc : 0;
        int gy = (ty0 - 2 + hy) & (DYc - 1);
        int gx = (tx0 - 2 + hx) & (DXc - 1);
        uint64_t gaddr = gbase +
            4ull * ((uint64_t)(((f * DZc + gz) * DYc + gy) * DXc + gx));
        uint32_t laddr = lds0 + 4u * (uint32_t)e;
        asm volatile("global_load_async_to_lds_b32 %0, %1, off"
                     :: "v"(laddr), "v"(gaddr) : "memory");
    }
    asm volatile("s_wait_asynccnt 0" ::: "memory");   // this wave's DMA done
    __syncthreads();                                  // all waves' halos in LDS

    // --- compute: each thread handles 2 cells of the 8^3 tile ---
    #pragma unroll 1
    for (int cc = 0; cc < 2; ++cc) {
        int c  = t + cc * 256;
        int lz = c >> 6;
        int ly = (c >> 3) & 7;
        int lx = c & 7;

        int n = idxg[((tz0 + lz) * DYc + (ty0 + ly)) * DXc + (tx0 + lx)];
        if (n < 0) continue;                          // empty cell: no particle

        int hb = ((lz + 2) * HS + (ly + 2)) * HS + (lx + 2);
        float cf[NF];
        #pragma unroll
        for (int f = 0; f < NF; ++f) cf[f] = tile[f * HC + hb];

        float a0 = 0.f, a1 = 0.f, a2 = 0.f, a3 = 0.f, a4 = 0.f, a5 = 0.f;

        // reference OFFSETS order: i outer, j, k inner; neighbor = cell - off
        #pragma unroll 1
        for (int i = 0; i < 5; ++i) {
            #pragma unroll 1
            for (int j = 0; j < 5; ++j) {
                // base index of the k=0 tap; k taps are 144 floats apart
                int nb0 = ((lz + 4) * HS + (ly + 4 - j)) * HS + (lx + 4 - i);
                #pragma unroll
                for (int k = 0; k < 5; ++k) {
                    int nbi = nb0 - k * (HS * HS);
                    float nbx  = tile[0 * HC + nbi];
                    float nby  = tile[1 * HC + nbi];
                    float nbz  = tile[2 * HC + nbi];
                    float nbvx = tile[3 * HC + nbi];
                    float nbvy = tile[4 * HC + nbi];
                    float nbvz = tile[5 * HC + nbi];
                    float nbwx = tile[6 * HC + nbi];
                    float nbwy = tile[7 * HC + nbi];
                    float nbwz = tile[8 * HC + nbi];

                    float dxp = cf[0] - nbx,  dyp = cf[1] - nby,  dzp = cf[2] - nbz;
                    float dvx = cf[3] - nbvx, dvy = cf[4] - nbvy, dvz = cf[5] - nbvz;
                    float wxs = cf[6] + nbwx, wys = cf[7] + nbwy, wzs = cf[8] + nbwz;

                    float dist = sqrtf(dxp * dxp + dyp * dyp + dzp * dzp);
                    float dd   = fmaxf(EPSc, dist);
                    float inv  = 1.0f / dd;
                    float nxn = dxp * inv, nyn = dyp * inv, nzn = dzp * inv;
                    float Vn  = dvx * nxn + dvy * nyn + dvz * nzn;
                    bool  contact = dist < two_d;
                    float fnc = contact ? KNc * (two_d - dist) : 0.0f;
                    float fnd = contact ? -eta * Vn : 0.0f;
                    float armx = Dc * nxn, army = Dc * nyn, armz = Dc * nzn;
                    float vtx = dvx - Vn * nxn + (wys * armz - wzs * army);
                    float vty = dvy - Vn * nyn + (wzs * armx - wxs * armz);
                    float vtz = dvz - Vn * nzn + (wxs * army - wys * armx);
                    float vt  = fmaxf(EPSc, sqrtf(vtx * vtx + vty * vty + vtz * vtz));
                    float ft  = (contact ? -MUc * (fabsf(fnc) + fabsf(fnd)) : 0.0f) / vt;
                    float ffx = ft * vtx, ffy = ft * vty, ffz = ft * vtz;
                    float fsum = fnc + fnd;
                    a0 += fsum * nxn + ffx;
                    a1 += fsum * nyn + ffy;
                    a2 += fsum * nzn + ffz;
                    a3 += army * ffz - armz * ffy;
                    a4 += armz * ffx - armx * ffz;
                    a5 += armx * ffy - army * ffx;
                }
            }
        }
        out[0 * N + n] = a0;
        out[1 * N + n] = a1;
        out[2 * N + n] = a2;
        out[3 * N + n] = a3;
        out[4 * N + n] = a4;
        out[5 * N + n] = a5;
    }
}

// ---------------------------------------------------------------------------
// Scatter kernel: apply mixer override, compute cell, scatter fields + index.
// ---------------------------------------------------------------------------
__global__ void __launch_bounds__(256) dem_build_grid(
    const float* __restrict__ x,  const float* __restrict__ y,  const float* __restrict__ z,
    const float* __restrict__ vx, const float* __restrict__ vy, const float* __restrict__ vz,
    const float* __restrict__ wx, const float* __restrict__ wy, const float* __restrict__ wz,
    const int* __restrict__ g3,
    float* __restrict__ grid, int* __restrict__ idxg,
    int N, float Dc, float mixc, float dt)
{
    int n = blockIdx.x * 256 + threadIdx.x;
    if (n >= N) return;

    float X = x[n], Y = y[n], Z = z[n];
    // cells come from the *pre-update* positions (reference order)
    int cx = (int)rintf(X / Dc);
    int cy = (int)rintf(Y / Dc);
    int cz = (int)rintf(Z / Dc);

    float VX = vx[n], VY = vy[n], VZ = vz[n];
    if (g3[n] != 0) {
        VX = -(Y - mixc);          // MIXER_W == 1.0
        VY =  (X - mixc);
        VZ = 0.0f;
        X = X + dt * VX;
        Y = Y + dt * VY;
        Z = Z + dt * VZ;
    }

    int cell = (cz * DYc + cy) * DXc + cx;
    grid[0 * NCELL + cell] = X;
    grid[1 * NCELL + cell] = Y;
    grid[2 * NCELL + cell] = Z;
    grid[3 * NCELL + cell] = VX;
    grid[4 * NCELL + cell] = VY;
    grid[5 * NCELL + cell] = VZ;
    grid[6 * NCELL + cell] = wx[n];
    grid[7 * NCELL + cell] = wy[n];
    grid[8 * NCELL + cell] = wz[n];
    idxg[cell] = n;
}

// ---------------------------------------------------------------------------
extern "C" void kernel_launch(void* const* d_in, const int* in_sizes, int n_in,
                              void* d_out, int out_size, void* d_ws, size_t ws_size,
                              hipStream_t stream)
{
    const float* x  = (const float*)d_in[0];
    const float* y  = (const float*)d_in[1];
    const float* z  = (const float*)d_in[2];
    const float* vx = (const float*)d_in[3];
    const float* vy = (const float*)d_in[4];
    const float* vz = (const float*)d_in[5];
    const float* wx = (const float*)d_in[6];
    const float* wy = (const float*)d_in[7];
    const float* wz = (const float*)d_in[8];
    const int*   g3 = (const int*)d_in[9];
    const int N = in_sizes[0];

    float* grid = (float*)d_ws;                                  // 9 * NCELL f32
    int*   idxg = (int*)((char*)d_ws + (size_t)NF * NCELL * 4);  // NCELL i32

    hipMemsetAsync(grid, 0,    (size_t)NF * NCELL * 4, stream);
    hipMemsetAsync(idxg, 0xFF, (size_t)NCELL * 4, stream);       // -1 = empty

    // reference constants, evaluated in double on host
    const double Dd    = 0.00054;
    const double KN    = 200.0;
    const double REST  = 0.3;
    const double alpha = -log(REST) / M_PI;
    const double gam   = alpha / sqrt(alpha * alpha + 1.0);
    const double mass  = 4.0 / 3.0 * 3.1415 * Dd * Dd * Dd * 2500.0;
    const float  eta   = (float)(2.0 * gam * sqrt(KN * mass / 2.0));
    const float  Dc    = (float)Dd;
    const float  two_d = (float)(2.0 * Dd);
    const float  mixc  = (float)(49.0 * Dd);      // (MIXER_R - 1) * D
    const float  dt    = (float)1e-5;

    dem_build_grid<<<(N + 255) / 256, 256, 0, stream>>>(
        x, y, z, vx, vy, vz, wx, wy, wz, g3, grid, idxg, N, Dc, mixc, dt);

    dem_forces<<<(DZc / TS) * (DYc / TS) * (DXc / TS), 256, 0, stream>>>(
        grid, idxg, (float*)d_out, N, Dc, two_d, eta);
}